// InterGCN_37967510897447
// MI455X (gfx1250) — compile-verified
//
#include <hip/hip_runtime.h>

typedef float v2f __attribute__((ext_vector_type(2)));
typedef float v8f __attribute__((ext_vector_type(8)));
typedef unsigned int v4u __attribute__((ext_vector_type(4)));
typedef int v4i __attribute__((ext_vector_type(4)));
typedef int v8i __attribute__((ext_vector_type(8)));

#define Bb   8
#define Sdim 64
#define Ndim 512
#define Cc   5
#define EPSF 1e-5f

// ---------------------------------------------------------------------------
// Kernel 1: one block per batch.
//   coef[c][i]  = aP[p]*lcW[p,i]            (p = pairidx(lc_i, c))
//   kterm[c][i] = aP[p]*lcB[p,i] + bP[p]
//   H[b,c,s]    = sum_i coef[c][i]*x[b,s,i] + kterm[c][i]
// Also copies lc through to the second output slot.
// ---------------------------------------------------------------------------
__global__ __launch_bounds__(320)
void intergcn_build_H(const float* __restrict__ x, const int* __restrict__ lc,
                      const float* __restrict__ aP, const float* __restrict__ bP,
                      const float* __restrict__ lcW, const float* __restrict__ lcB,
                      float* __restrict__ Hws, int* __restrict__ lc_out) {
    const int b = blockIdx.x;
    const int t = threadIdx.x;                 // 320 threads = 10 waves

    __shared__ float coef[Cc][Ndim];           // 10 KB
    __shared__ float kterm[Cc][Ndim];          // 10 KB

    for (int i = t; i < Ndim; i += 320) {
        const int li = lc[b * Ndim + i];
        lc_out[b * Ndim + i] = li;             // pass-through output
#pragma unroll
        for (int c = 0; c < Cc; ++c) {
            const int mx = li > c ? li : c;
            const int mn = li > c ? c : li;
            const int p  = (mx * (mx + 1)) / 2 + mn;
            const float a = aP[p];
            coef[c][i]  = a * lcW[p * Ndim + i];
            kterm[c][i] = __builtin_fmaf(a, lcB[p * Ndim + i], bP[p]);
        }
    }
    __syncthreads();

    const int c = t >> 6;                      // 0..4
    const int s = t & 63;                      // 0..63
    const float* xr = &x[(b * Sdim + s) * Ndim];
    float acc = 0.f;
#pragma unroll 8
    for (int i = 0; i < Ndim; ++i)
        acc += __builtin_fmaf(coef[c][i], xr[i], kterm[c][i]);
    Hws[(b * Cc + c) * Sdim + s] = acc;
}

// ---------------------------------------------------------------------------
// Kernel 2: one block per (batch, 128-node tile). 256 threads = 8 waves.
// TDM DMAs W2 (64x64 f32) into LDS with hardware padding (row stride 66)
// while the VALU builds the R tile; then 64x64 x 64x128 GEMM via
// V_WMMA_F32_16X16X4_F32 with bn2 folded into the store.
// ---------------------------------------------------------------------------
#define WSTR 66    // 64 data DWORDs + 2 pad DWORDs per row (TDM pad_amount=1)
#define RSTR 132   // padded row stride for R in LDS

__global__ __launch_bounds__(256)
void intergcn_gemm(const float* __restrict__ x, const int* __restrict__ lc,
                   const float* __restrict__ w2,
                   const float* __restrict__ g1, const float* __restrict__ be1,
                   const float* __restrict__ rm1, const float* __restrict__ rv1,
                   const float* __restrict__ g2, const float* __restrict__ be2,
                   const float* __restrict__ rm2, const float* __restrict__ rv2,
                   const float* __restrict__ Hws, float* __restrict__ out) {
    const int b  = blockIdx.x >> 2;
    const int n0 = (blockIdx.x & 3) * 128;
    const int t  = threadIdx.x;

    __shared__ float sW2[Sdim * WSTR];         // 16.9 KB (filled by TDM)
    __shared__ float sR[Sdim * RSTR];          // 33.8 KB
    __shared__ float sH[Cc * Sdim];            // 1.3 KB
    __shared__ float sMul1[128], sAdd1[128];
    __shared__ float sS2[Sdim], sAdd2[Sdim];
    __shared__ int   sLc[128];

    // ---- issue TDM copy of W2 -> sW2 (wave 0 only; overlaps R-tile build) ----
    if (t < 32) {
        const unsigned lds_off = (unsigned)(size_t)(&sW2[0]); // low 32b = LDS offset
        const unsigned long long ga = (unsigned long long)(size_t)w2;
        v4u g0;
        g0.x = 1u;                                  // count=1, user mode, no gather
        g0.y = lds_off;                             // lds_addr (bytes)
        g0.z = (unsigned)(ga & 0xFFFFFFFFull);      // global_addr[31:0]
        g0.w = (unsigned)((ga >> 32) & 0x1FFFFFFull) | 0x80000000u; // addr[56:32] | type=2
        v8i g1d;
        g1d[0] = (int)((2u << 16)      |            // data_size = 4B
                       (1u << 20)      |            // pad_enable
                       (5u << 22)      |            // pad_interval: 64 DWORDs
                       (1u << 25));                 // pad_amount: 2 DWORDs
        g1d[1] = (int)(64u << 16);                  // tensor_dim0 = 64
        g1d[2] = (int)(64u << 16);                  // tensor_dim1 = 64
        g1d[3] = (int)(64u << 16);                  // tile_dim0 = 64
        g1d[4] = 64;                                // tile_dim1 = 64, tile_dim2 = 0
        g1d[5] = 64;                                // tensor_dim0_stride = 64
        g1d[6] = 0;                                 // tensor_dim1_stride = 0
        g1d[7] = 0;
        v4i z4 = {0, 0, 0, 0};
#if __clang_major__ >= 23
        v8i z8 = {0, 0, 0, 0, 0, 0, 0, 0};
        __builtin_amdgcn_tensor_load_to_lds(g0, g1d, z4, z4, z8, 0);
#else
        __builtin_amdgcn_tensor_load_to_lds(g0, g1d, z4, z4, 0);
#endif
    }

    // ---- stage small parameters while the TDM runs ----
    for (int i = t; i < Cc * Sdim; i += 256) sH[i] = Hws[b * Cc * Sdim + i];
    if (t < 128) {
        const int gn = n0 + t;
        const float s1 = g1[gn] * rsqrtf(rv1[gn] + EPSF);
        sMul1[t] = s1;
        sAdd1[t] = __builtin_fmaf(-rm1[gn], s1, be1[gn]);
        sLc[t]   = lc[b * Ndim + gn];
    }
    if (t < Sdim) {
        const float s2 = g2[t] * rsqrtf(rv2[t] + EPSF);
        sS2[t]   = s2;
        sAdd2[t] = __builtin_fmaf(-rm2[t], s2, be2[t]);
    }
    __syncthreads();

    // ---- bn1 + residual + relu into LDS tile r[s][n] (float4 x loads) ----
    {
        const int cg = t & 31;                 // column group: 4 nodes each
        const int sr = t >> 5;                 // starting row, stride 8
        const int nl = cg * 4;
        const float4* xp = (const float4*)(x + (size_t)b * Sdim * Ndim + n0 + nl);
        const float m10 = sMul1[nl + 0], a10 = sAdd1[nl + 0];
        const float m11 = sMul1[nl + 1], a11 = sAdd1[nl + 1];
        const float m12 = sMul1[nl + 2], a12 = sAdd1[nl + 2];
        const float m13 = sMul1[nl + 3], a13 = sAdd1[nl + 3];
        const float* h0 = &sH[sLc[nl + 0] * Sdim];
        const float* h1 = &sH[sLc[nl + 1] * Sdim];
        const float* h2 = &sH[sLc[nl + 2] * Sdim];
        const float* h3 = &sH[sLc[nl + 3] * Sdim];
#pragma unroll
        for (int s = sr; s < Sdim; s += 8) {
            const float4 xv = xp[(size_t)s * (Ndim / 4)];
            float4 r;
            r.x = __builtin_fmaf(h0[s], m10, a10) + xv.x;
            r.y = __builtin_fmaf(h1[s], m11, a11) + xv.y;
            r.z = __builtin_fmaf(h2[s], m12, a12) + xv.z;
            r.w = __builtin_fmaf(h3[s], m13, a13) + xv.w;
            r.x = r.x > 0.f ? r.x : 0.f;
            r.y = r.y > 0.f ? r.y : 0.f;
            r.z = r.z > 0.f ? r.z : 0.f;
            r.w = r.w > 0.f ? r.w : 0.f;
            *(float4*)(&sR[s * RSTR + nl]) = r;
        }
    }

    // ---- make W2 DMA visible, then release all waves into the GEMM ----
    if (t < 32) __builtin_amdgcn_s_wait_tensorcnt(0);
    __syncthreads();

    // ---- WMMA GEMM: out[o,n] = sum_s W2[o,s] * r[s,n] ----
    const int wave = t >> 5;                   // 0..7 -> N tile
    const int lane = t & 31;
    const int m    = lane & 15;
    const int half = lane >> 4;                // lanes 16-31 carry K+2 / M+8
    const int no   = wave * 16;

    v8f acc[4];
#pragma unroll
    for (int q = 0; q < 4; ++q) acc[q] = (v8f){0.f,0.f,0.f,0.f,0.f,0.f,0.f,0.f};

#pragma unroll
    for (int kk = 0; kk < Sdim; kk += 4) {
        const int kb = kk + 2 * half;
        // B fragment (4x16 of r), shared by all 4 M tiles of this wave
        v2f bf = { sR[kb * RSTR + no + m], sR[(kb + 1) * RSTR + no + m] };
#pragma unroll
        for (int q = 0; q < 4; ++q) {
            // A fragment (16x4 of W2)
            v2f af = { sW2[(q * 16 + m) * WSTR + kb],
                       sW2[(q * 16 + m) * WSTR + kb + 1] };
            acc[q] = __builtin_amdgcn_wmma_f32_16x16x4_f32(
                         false, af, false, bf, (short)0, acc[q], false, false);
        }
    }

    // ---- bn2 + store ----
    const int gn = n0 + no + m;
#pragma unroll
    for (int q = 0; q < 4; ++q) {
#pragma unroll
        for (int rr = 0; rr < 8; ++rr) {
            const int o = q * 16 + half * 8 + rr;
            out[(b * Sdim + o) * Ndim + gn] =
                __builtin_fmaf(acc[q][rr], sS2[o], sAdd2[o]);
        }
    }
}

// ---------------------------------------------------------------------------
extern "C" void kernel_launch(void* const* d_in, const int* in_sizes, int n_in,
                              void* d_out, int out_size, void* d_ws, size_t ws_size,
                              hipStream_t stream) {
    const float* x   = (const float*)d_in[0];
    const int*   lc  = (const int*)d_in[1];
    const float* aP  = (const float*)d_in[2];
    const float* bP  = (const float*)d_in[3];
    const float* lcW = (const float*)d_in[4];
    const float* lcB = (const float*)d_in[5];
    const float* w2  = (const float*)d_in[6];
    const float* g1  = (const float*)d_in[7];
    const float* be1 = (const float*)d_in[8];
    const float* rm1 = (const float*)d_in[9];
    const float* rv1 = (const float*)d_in[10];
    const float* g2  = (const float*)d_in[11];
    const float* be2 = (const float*)d_in[12];
    const float* rm2 = (const float*)d_in[13];
    const float* rv2 = (const float*)d_in[14];

    float* out    = (float*)d_out;
    float* Hws    = (float*)d_ws;                       // B*C*S floats = 10 KB
    int*   lc_out = (int*)(out + Bb * Sdim * Ndim);     // second tuple output

    intergcn_build_H<<<Bb, 320, 0, stream>>>(x, lc, aP, bP, lcW, lcB, Hws, lc_out);
    intergcn_gemm<<<Bb * 4, 256, 0, stream>>>(x, lc, w2, g1, be1, rm1, rv1,
                                              g2, be2, rm2, rv2, Hws, out);
}